// GraphDiscriminator_18391049961795
// MI455X (gfx1250) — compile-verified
//
#include <hip/hip_runtime.h>
#include <hip/hip_bf16.h>

typedef __attribute__((ext_vector_type(2))) float v2f;
typedef __attribute__((ext_vector_type(8))) float v8f;

#define FEAT 512
#define HID  16
#define NGRAPH 64

// -------------------------------------------------------------------------
// Kernel 1: h = x @ W  via V_WMMA_F32_16X16X4_F32.
// One wave -> one 16-row x 16-col C tile. K=512 stepped by 4.
// Block = 64 threads (2 waves). Grid = N/32 blocks (N % 32 == 0 for N=100000).
// -------------------------------------------------------------------------
__global__ void __launch_bounds__(64) gcn_gemm_kernel(const float* __restrict__ x,
                                                      const float* __restrict__ W,
                                                      float* __restrict__ h) {
    __shared__ float lw[FEAT * HID];  // 32 KB: whole W matrix
    for (int i = threadIdx.x; i < FEAT * HID; i += 64) lw[i] = W[i];
    __syncthreads();

    const int lane = threadIdx.x & 31;
    const int wave = threadIdx.x >> 5;
    const int tile = blockIdx.x * 2 + wave;
    const int row_base = tile * 16;

    const int m    = lane & 15;        // row within tile (A) / column N (B,C)
    const int koff = (lane >> 4) * 2;  // lanes 16-31 hold K+2,K+3

    const float* __restrict__ xrow = x + (size_t)(row_base + m) * FEAT;

    v8f c = {};
#pragma unroll 4
    for (int k = 0; k < FEAT; k += 4) {
        v2f a, b;
        // A 16x4 f32 layout: lane m, VGPR0 = K=koff, VGPR1 = K=koff+1
        a.x = xrow[k + koff];
        a.y = xrow[k + koff + 1];
        // B 4x16 f32 layout: VGPR0 lanes0-15 = row K, lanes16-31 = row K+2
        b.x = lw[(k + koff) * HID + m];
        b.y = lw[(k + koff + 1) * HID + m];
        c = __builtin_amdgcn_wmma_f32_16x16x4_f32(
            /*neg_a=*/false, a, /*neg_b=*/false, b,
            /*c_mod=*/(short)0, c, /*reuse_a=*/false, /*reuse_b=*/false);
    }

    // C 16x16 f32 layout: VGPR r -> M=r (lanes 0-15), M=r+8 (lanes 16-31); N = m
    const int rowadd = (lane >> 4) * 8;
#pragma unroll
    for (int r = 0; r < 8; ++r) {
        h[(size_t)(row_base + r + rowadd) * HID + m] = c[r];
    }
}

// -------------------------------------------------------------------------
// Degree / normalization
// -------------------------------------------------------------------------
__global__ void deg_init_kernel(float* __restrict__ deg, int n) {
    int i = blockIdx.x * blockDim.x + threadIdx.x;
    if (i < n) deg[i] = 1.0f;  // self-loop
}

__global__ void deg_edges_kernel(const int* __restrict__ dst, float* __restrict__ deg, int E) {
    int e = blockIdx.x * blockDim.x + threadIdx.x;
    if (e < E) atomicAdd(&deg[dst[e]], 1.0f);
}

__global__ void dinv_kernel(float* __restrict__ deg, int n) {
    int i = blockIdx.x * blockDim.x + threadIdx.x;
    if (i < n) {
        float d = deg[i];                 // always >= 1 (self-loops)
        deg[i] = 1.0f / sqrtf(d);         // full-precision rsqrt
    }
}

// -------------------------------------------------------------------------
// Aggregation: self-loop init then edge scatter-add
// -------------------------------------------------------------------------
__global__ void agg_init_kernel(const float* __restrict__ h, const float* __restrict__ dinv,
                                float* __restrict__ agg, int n16) {
    int t = blockIdx.x * blockDim.x + threadIdx.x;
    if (t < n16) {
        int i = t >> 4;
        float di = dinv[i];
        agg[t] = h[t] * di * di;  // self-loop message, norm = dinv[i]^2
    }
}

__global__ void agg_edges_kernel(const int* __restrict__ ei, const float* __restrict__ h,
                                 const float* __restrict__ dinv, float* __restrict__ agg, int E) {
    int t = blockIdx.x * blockDim.x + threadIdx.x;
    if (t >= E * HID) return;
    int e = t >> 4;
    int f = t & 15;
    int s = ei[e];
    int d = ei[E + e];
    float norm = dinv[s] * dinv[d];
    atomicAdd(&agg[(size_t)d * HID + f], h[(size_t)s * HID + f] * norm);
}

// -------------------------------------------------------------------------
// Pooling: bias + relu fused, LDS partial sums per block
// -------------------------------------------------------------------------
__global__ void zero_kernel(float* __restrict__ p, int n) {
    int i = blockIdx.x * blockDim.x + threadIdx.x;
    if (i < n) p[i] = 0.0f;
}

__global__ void __launch_bounds__(256) pool_kernel(const float* __restrict__ agg,
                                                   const float* __restrict__ b_conv,
                                                   const int* __restrict__ batch,
                                                   float* __restrict__ sums,
                                                   float* __restrict__ counts, int n) {
    __shared__ float ls[NGRAPH * HID];
    __shared__ float lc[NGRAPH];
    for (int i = threadIdx.x; i < NGRAPH * HID; i += 256) ls[i] = 0.0f;
    if (threadIdx.x < NGRAPH) lc[threadIdx.x] = 0.0f;
    __syncthreads();

    const int total = n * HID;
    const int stride = gridDim.x * 256;
    for (int t = blockIdx.x * 256 + threadIdx.x; t < total; t += stride) {
        int i = t >> 4;
        int f = t & 15;
        int g = batch[i];
        float v = agg[t] + b_conv[f];
        v = fmaxf(v, 0.0f);
        atomicAdd(&ls[g * HID + f], v);
        if (f == 0) atomicAdd(&lc[g], 1.0f);
    }
    __syncthreads();

    for (int i = threadIdx.x; i < NGRAPH * HID; i += 256)
        if (ls[i] != 0.0f) atomicAdd(&sums[i], ls[i]);
    if (threadIdx.x < NGRAPH && lc[threadIdx.x] != 0.0f)
        atomicAdd(&counts[threadIdx.x], lc[threadIdx.x]);
}

// -------------------------------------------------------------------------
// Final classifier: logits[g][c] = (sums[g]/cnt) @ W_cls + b_cls
// -------------------------------------------------------------------------
__global__ void logits_kernel(const float* __restrict__ sums, const float* __restrict__ counts,
                              const float* __restrict__ W_cls, const float* __restrict__ b_cls,
                              float* __restrict__ out) {
    int t = threadIdx.x;
    if (t >= NGRAPH * 2) return;
    int g = t >> 1;
    int c = t & 1;
    float cnt = fmaxf(counts[g], 1.0f);
    float acc = b_cls[c];
#pragma unroll
    for (int f = 0; f < HID; ++f)
        acc += (sums[g * HID + f] / cnt) * W_cls[f * 2 + c];
    out[g * 2 + c] = acc;
}

// -------------------------------------------------------------------------
// Launcher
// -------------------------------------------------------------------------
extern "C" void kernel_launch(void* const* d_in, const int* in_sizes, int n_in,
                              void* d_out, int out_size, void* d_ws, size_t ws_size,
                              hipStream_t stream) {
    const float* x      = (const float*)d_in[0];
    const float* W      = (const float*)d_in[1];
    const float* b_conv = (const float*)d_in[2];
    const float* W_cls  = (const float*)d_in[3];
    const float* b_cls  = (const float*)d_in[4];
    const int*   ei     = (const int*)d_in[5];   // [2, E]: src then dst
    const int*   batch  = (const int*)d_in[6];

    const int N = in_sizes[0] / FEAT;   // 100000
    const int E = in_sizes[5] / 2;      // 3200000

    // Workspace layout
    char* ws = (char*)d_ws;
    float* h      = (float*)ws;                         ws += (size_t)N * HID * sizeof(float);
    float* agg    = (float*)ws;                         ws += (size_t)N * HID * sizeof(float);
    float* dinv   = (float*)ws;                         ws += (size_t)N * sizeof(float);
    float* sums   = (float*)ws;                         ws += NGRAPH * HID * sizeof(float);
    float* counts = (float*)ws;

    // 1. GEMM h = x @ W  (WMMA path). N=100000 is a multiple of 32.
    gcn_gemm_kernel<<<N / 32, 64, 0, stream>>>(x, W, h);

    // 2. Degrees (self-loop = 1) + edge increments, then dinv = 1/sqrt(deg)
    deg_init_kernel<<<(N + 255) / 256, 256, 0, stream>>>(dinv, N);
    deg_edges_kernel<<<(E + 255) / 256, 256, 0, stream>>>(ei + E, dinv, E);
    dinv_kernel<<<(N + 255) / 256, 256, 0, stream>>>(dinv, N);

    // 3. Aggregation: self-loop init, then edge scatter (1 thread per edge-feature)
    agg_init_kernel<<<((size_t)N * HID + 255) / 256, 256, 0, stream>>>(h, dinv, agg, N * HID);
    {
        long long work = (long long)E * HID;
        int blocks = (int)((work + 255) / 256);
        agg_edges_kernel<<<blocks, 256, 0, stream>>>(ei, h, dinv, agg, E);
    }

    // 4. Pooling (bias + relu fused), with zero-init of accumulators each call
    zero_kernel<<<(NGRAPH * HID + NGRAPH + 255) / 256, 256, 0, stream>>>(sums, NGRAPH * HID + NGRAPH);
    pool_kernel<<<512, 256, 0, stream>>>(agg, b_conv, batch, sums, counts, N);

    // 5. Classifier
    logits_kernel<<<1, 128, 0, stream>>>(sums, counts, W_cls, b_cls, (float*)d_out);
}